// BlenderbotSelfAttention_57509612093539
// MI455X (gfx1250) — compile-verified
//
#include <hip/hip_runtime.h>
#include <math.h>

typedef __attribute__((ext_vector_type(2))) float v2f;
typedef __attribute__((ext_vector_type(4))) float v4f;
typedef __attribute__((ext_vector_type(8))) float v8f;

#define DMODEL 2048
#define NHEAD 32
#define HEAD_DIM 64
#define SKV 4096
#define BATCH 16

// ---------------------------------------------------------------------------
// Y[16 x 2048] = X[16 x 2048] @ W[2048 x 2048] + bias
// One wave = one 16x16 tile via V_WMMA_F32_16X16X4_F32, 4 waves/block -> 64 cols.
// Software-pipelined: fragments for k+8/k+12 are loaded before the WMMAs that
// consume k/k+4, so loads overlap matrix ops instead of serializing on
// s_wait_loadcnt. Prefetch is unconditional (speculative TH: OOB silently
// dropped per ISA), avoiding per-iteration branches.
// ---------------------------------------------------------------------------
__global__ __launch_bounds__(128) void gemm16_wmma(const float* __restrict__ X,
                                                   const float* __restrict__ W,
                                                   const float* __restrict__ bias,
                                                   float* __restrict__ Y) {
  const int lane = threadIdx.x & 31;
  const int wave = threadIdx.x >> 5;
  const int m    = lane & 15;
  const int kk   = (lane >> 4) << 1;              // 0 or 2
  const int ncol = blockIdx.x * 64 + wave * 16 + (lane & 15);
  const float* xrow = X + (size_t)m * DMODEL;

  v2f a0, b0, a1, b1;
  // Preload fragments for k=0 and k=4.
  a0.x = xrow[kk];
  a0.y = xrow[kk + 1];
  b0.x = W[(size_t)kk * DMODEL + ncol];
  b0.y = W[(size_t)(kk + 1) * DMODEL + ncol];
  a1.x = xrow[4 + kk];
  a1.y = xrow[4 + kk + 1];
  b1.x = W[(size_t)(4 + kk) * DMODEL + ncol];
  b1.y = W[(size_t)(4 + kk + 1) * DMODEL + ncol];

  v8f c = {};
  int k = 0;
  for (; k < DMODEL - 8; k += 8) {
    // Speculative prefetch ~256 rows ahead of the weight stream.
    __builtin_prefetch(&W[(size_t)(k + 256 + kk) * DMODEL + ncol], 0, 1);

    // Load k+8 fragment, consume k fragment.
    v2f na, nb;
    na.x = xrow[k + 8 + kk];
    na.y = xrow[k + 8 + kk + 1];
    nb.x = W[(size_t)(k + 8 + kk) * DMODEL + ncol];
    nb.y = W[(size_t)(k + 8 + kk + 1) * DMODEL + ncol];
    c = __builtin_amdgcn_wmma_f32_16x16x4_f32(false, a0, false, b0, (short)0, c,
                                              false, false);
    a0 = na; b0 = nb;

    // Load k+12 fragment, consume k+4 fragment.
    na.x = xrow[k + 12 + kk];
    na.y = xrow[k + 12 + kk + 1];
    nb.x = W[(size_t)(k + 12 + kk) * DMODEL + ncol];
    nb.y = W[(size_t)(k + 12 + kk + 1) * DMODEL + ncol];
    c = __builtin_amdgcn_wmma_f32_16x16x4_f32(false, a1, false, b1, (short)0, c,
                                              false, false);
    a1 = na; b1 = nb;
  }
  // Tail: fragments for k = DMODEL-8 and DMODEL-4 already in registers.
  c = __builtin_amdgcn_wmma_f32_16x16x4_f32(false, a0, false, b0, (short)0, c,
                                            false, false);
  c = __builtin_amdgcn_wmma_f32_16x16x4_f32(false, a1, false, b1, (short)0, c,
                                            false, false);

  const float bn    = bias[ncol];
  const int   mbase = (lane >> 4) << 3;           // 0 or 8
#pragma unroll
  for (int r = 0; r < 8; ++r)
    Y[(size_t)(mbase + r) * DMODEL + ncol] = c[r] + bn;
}

// ---------------------------------------------------------------------------
// Fused per-(b,h): stream K/V slice from input cache -> output cache once
// (NT hints: 1 GB stream, no reuse), substituting row last_pos[b] with the
// fresh k/v, while computing masked softmax(q.K/8) @ V into attn_out.
// ---------------------------------------------------------------------------
__global__ __launch_bounds__(256) void attn_fused(const float* __restrict__ qn,
                                                  const float* __restrict__ kn,
                                                  const float* __restrict__ vn,
                                                  const int* __restrict__ last_pos,
                                                  const float* __restrict__ cacheK,
                                                  const float* __restrict__ cacheV,
                                                  float* __restrict__ outK,
                                                  float* __restrict__ outV,
                                                  float* __restrict__ attn_out) {
  __shared__ float qs[HEAD_DIM];
  __shared__ float knew[HEAD_DIM];
  __shared__ float vnew[HEAD_DIM];
  __shared__ float sc[SKV];
  __shared__ float red[256];
  __shared__ float part[4][HEAD_DIM];

  const int bh  = blockIdx.x;            // 0..511
  const int b   = bh >> 5;
  const int h   = bh & 31;
  const int tid = threadIdx.x;
  const size_t hoff = (size_t)b * DMODEL + (size_t)h * HEAD_DIM;

  if (tid < HEAD_DIM) {
    qs[tid]   = qn[hoff + tid];
    knew[tid] = kn[hoff + tid];
    vnew[tid] = vn[hoff + tid];
  }
  __syncthreads();

  const int lp     = last_pos[b];
  const int nvalid = lp + 1;
  const size_t base = (size_t)bh * SKV * HEAD_DIM;
  const float* Kb = cacheK + base;
  const float* Vb = cacheV + base;
  float*       Ko = outK + base;
  float*       Vo = outV + base;
  const float scale = 0.125f;            // 1/sqrt(64)

  // Phase 1: stream K rows -> output cache, dot with q.
  float lmax = -3.0e38f;
  for (int s = tid; s < SKV; s += 256) {
    const v4f* kr = (const v4f*)(Kb + (size_t)s * HEAD_DIM);
    v4f*       kw = (v4f*)(Ko + (size_t)s * HEAD_DIM);
    const bool isnew = (s == lp);
    float acc = 0.f;
#pragma unroll
    for (int d4 = 0; d4 < 16; ++d4) {
      v4f kv;
      if (isnew) {
        kv.x = knew[4 * d4 + 0]; kv.y = knew[4 * d4 + 1];
        kv.z = knew[4 * d4 + 2]; kv.w = knew[4 * d4 + 3];
      } else {
        kv = __builtin_nontemporal_load(&kr[d4]);
      }
      __builtin_nontemporal_store(kv, &kw[d4]);
      acc = fmaf(qs[4 * d4 + 0], kv.x, acc);
      acc = fmaf(qs[4 * d4 + 1], kv.y, acc);
      acc = fmaf(qs[4 * d4 + 2], kv.z, acc);
      acc = fmaf(qs[4 * d4 + 3], kv.w, acc);
    }
    const float sval = (s < nvalid) ? acc * scale : -3.0e38f;
    sc[s] = sval;
    lmax = fmaxf(lmax, sval);
  }

  // Workgroup max-reduce.
  red[tid] = lmax;
  __syncthreads();
  for (int off = 128; off > 0; off >>= 1) {
    if (tid < off) red[tid] = fmaxf(red[tid], red[tid + off]);
    __syncthreads();
  }
  const float mx = red[0];
  __syncthreads();

  // Phase 2: exp + sum.
  float lsum = 0.f;
  for (int s = tid; s < SKV; s += 256) {
    const float p = (s < nvalid) ? __expf(sc[s] - mx) : 0.f;
    sc[s] = p;
    lsum += p;
  }
  red[tid] = lsum;
  __syncthreads();
  for (int off = 128; off > 0; off >>= 1) {
    if (tid < off) red[tid] += red[tid + off];
    __syncthreads();
  }
  const float inv = 1.0f / red[0];

  // Phase 3: stream V rows -> output cache, weighted accumulate.
  // tid -> (d = tid&63, group g = tid>>6); each (s,d) touched exactly once.
  const int d = tid & 63;
  const int g = tid >> 6;
  float acc = 0.f;
  for (int s = g; s < SKV; s += 4) {
    const float val = (s == lp) ? vnew[d]
                                : __builtin_nontemporal_load(&Vb[(size_t)s * HEAD_DIM + d]);
    __builtin_nontemporal_store(val, &Vo[(size_t)s * HEAD_DIM + d]);
    acc = fmaf(sc[s], val, acc);
  }
  part[g][d] = acc;
  __syncthreads();

  if (tid < HEAD_DIM)
    attn_out[hoff + tid] =
        (part[0][tid] + part[1][tid] + part[2][tid] + part[3][tid]) * inv;
}

// ---------------------------------------------------------------------------
extern "C" void kernel_launch(void* const* d_in, const int* in_sizes, int n_in,
                              void* d_out, int out_size, void* d_ws, size_t ws_size,
                              hipStream_t stream) {
  const float* x        = (const float*)d_in[0];
  const int*   last_pos = (const int*)d_in[1];
  // d_in[2] = mask (recomputed from last_pos; unused)
  const float* Wq = (const float*)d_in[3];
  const float* bq = (const float*)d_in[4];
  const float* Wk = (const float*)d_in[5];
  const float* bk = (const float*)d_in[6];
  const float* Wv = (const float*)d_in[7];
  const float* bv = (const float*)d_in[8];
  const float* Wo = (const float*)d_in[9];
  const float* bo = (const float*)d_in[10];
  const float* cacheK = (const float*)d_in[11];
  const float* cacheV = (const float*)d_in[12];

  const size_t cache_elems = (size_t)BATCH * NHEAD * SKV * HEAD_DIM; // 134217728
  float* out  = (float*)d_out;                 // [16, 2048]
  float* outK = out + (size_t)BATCH * DMODEL;  // updated_kcache
  float* outV = outK + cache_elems;            // updated_vcache

  float* q    = (float*)d_ws;                  // [16, 2048]
  float* k    = q + (size_t)BATCH * DMODEL;
  float* v    = k + (size_t)BATCH * DMODEL;
  float* attn = v + (size_t)BATCH * DMODEL;

  // QKV projections (WMMA f32 GEMMs).
  gemm16_wmma<<<32, 128, 0, stream>>>(x, Wq, bq, q);
  gemm16_wmma<<<32, 128, 0, stream>>>(x, Wk, bk, k);
  gemm16_wmma<<<32, 128, 0, stream>>>(x, Wv, bv, v);

  // Fused cache copy + scatter-update + masked softmax attention.
  attn_fused<<<BATCH * NHEAD, 256, 0, stream>>>(q, k, v, last_pos, cacheK, cacheV,
                                                outK, outV, attn);

  // Output projection.
  gemm16_wmma<<<32, 128, 0, stream>>>(attn, Wo, bo, out);
}